// Forward_Model_27616639714077
// MI455X (gfx1250) — compile-verified
//
#include <hip/hip_runtime.h>
#include <hip/hip_bf16.h>
#include <math.h>

// MI455X (gfx1250) fused Sobel-magnitude depthwise 3x3 kernel, TDM edition.
//
// Roofline: 64 MiB in + 64 MiB out at 23.3 TB/s => ~5.6 us; ~0.65 GFLOP of
// f32 => bandwidth-bound (5 FLOP/byte). Data movement uses the CDNA5 Tensor
// Data Mover: one tensor_load_to_lds per block DMAs the (32+2-halo) x 256 f32
// tile (2-D strided) into LDS; hardware OOB-zero (tensor_dim1) supplies the
// bottom zero-padding row. WMMA is deliberately NOT used: a depthwise 3x3
// with one shared scalar kernel maps to <=12.5% tensor utilization via
// im2col, and the op is nowhere near compute-bound.
//
// Compute: thread (rs,cg) = rows [8rs,8rs+8) x cols [4cg,4cg+4). Per output:
// 15 FMA (the 3 diagonal taps of conv(k) and conv(k^T) are shared), 0.75 DS
// loads (one ds_load_b128 + 2 scalars per 4 outputs), 0.25 global_store_b128,
// 1 v_add_f32 with |a|+|b| source modifiers.

#define TH     32            // output rows per block
#define WROWS  (TH + 2)      // staged rows incl. halo
#define WIDTH  256
#define NCH    256
#define NROWS  256

typedef __attribute__((ext_vector_type(4))) unsigned u32x4;
typedef __attribute__((ext_vector_type(8))) unsigned u32x8;

__global__ __launch_bounds__(256) void sobel_g_kernel(
    const float* __restrict__ x,     // [256,256,256]
    const float* __restrict__ w,     // [1,9]
    const float* __restrict__ wf,    // [1,1]
    float* __restrict__ out)         // [1,1,256,256,256]
{
    __shared__ float tile[WROWS * WIDTH];

    const int tid  = threadIdx.x;          // 0..255
    const int ch   = blockIdx.y;           // channel
    const int row0 = blockIdx.x * TH;      // first output row of this tile

    // Flat pointer to LDS truncates to the 32-bit LDS byte offset
    // (ISA: LDS aperture address calc uses addr[31:0]).
    const unsigned ldsBase = (unsigned)(unsigned long long)(void*)tile;

    // Top tile: global row -1 is not addressable; skip LDS row 0 in the DMA
    // and zero-fill it with DS stores instead. Bottom tile: tensor_dim1 =
    // remaining valid rows, so the requested 34th row arrives as zeros.
    const int topTile   = (row0 == 0);
    const int grStart   = topTile ? 0 : (row0 - 1);     // first DMA'd row
    const int nrows     = WROWS - topTile;              // 33 or 34 tile rows
    const unsigned td1  = (unsigned)(NROWS - grStart);  // valid rows from start

    if (topTile) tile[tid] = 0.f;   // zero halo row 0 (256 threads x 1 float)

    // Only one wave per block may issue the DMA (TDM ignores EXEC, so this
    // must be a *scalar* branch or all 8 waves would replicate the copy).
    const unsigned wid = __builtin_amdgcn_readfirstlane((unsigned)tid >> 5);
    if (wid == 0) {
        unsigned long long ga = (unsigned long long)x
                              + ((unsigned long long)ch      << 18)   // ch*256*256*4
                              + ((unsigned long long)grStart << 10);  // row*256*4

        // ---- Tensor DMA Descriptor (ISA cdna5 ch.8), groups 0+1 (2-D) ----
        u32x4 g0;
        g0.x = 1u;                                         // count=1, no gather
        g0.y = ldsBase + (topTile ? (unsigned)(WIDTH * 4) : 0u); // lds_addr
        g0.z = (unsigned)ga;                               // global_addr[31:0]
        g0.w = ((unsigned)(ga >> 32) & 0x01FFFFFFu)        // global_addr[56:32]
             | (2u << 30);                                 // type=2 ("image")

        u32x8 g1;
        g1.s0 = (2u << 16);                  // data_size=2 -> 4B elements
        g1.s1 = ((unsigned)WIDTH & 0xFFFFu) << 16;         // tensor_dim0 lo16
        g1.s2 = (((unsigned)WIDTH >> 16) & 0xFFFFu)        // tensor_dim0 hi16
              | ((td1 & 0xFFFFu) << 16);                   // tensor_dim1 lo16
        g1.s3 = ((td1 >> 16) & 0xFFFFu)                    // tensor_dim1 hi16
              | ((unsigned)WIDTH << 16);                   // tile_dim0 = 256
        g1.s4 = (unsigned)nrows;             // tile_dim1; tile_dim2=0 (2-D)
        g1.s5 = (unsigned)WIDTH;             // tensor_dim0_stride[31:0] = 256
        g1.s6 = 0u;                          // stride0 hi16 | stride1 lo16
        g1.s7 = 0u;                          // stride1 hi32 (unused for 2-D)

        // 2-group form (tensors up to 2-D): VADDR2/3 = NULL.
        asm volatile("tensor_load_to_lds %0, %1"
                     :
                     : "s"(g0), "s"(g1)
                     : "memory");
        __builtin_amdgcn_s_wait_tensorcnt(0);   // drain TENSORcnt
    }
    __syncthreads();   // publish LDS tile (and the DS zero row) to all waves

    // ---- Build the clamped, scaled 3x3 kernel (uniform -> scalarized) ----
    const float f = fminf(fmaxf(wf[0], 1.001f), 254.999f);
    float k0 = fminf(fmaxf(w[0], -1.f), 1.f) * f;
    float k1 = fminf(fmaxf(w[1], -1.f), 1.f) * f;
    float k2 = fminf(fmaxf(w[2], -1.f), 1.f) * f;
    float k3 = fminf(fmaxf(w[3], -1.f), 1.f) * f;
    float k4 = fminf(fmaxf(w[4], -1.f), 1.f) * f;
    float k5 = fminf(fmaxf(w[5], -1.f), 1.f) * f;
    float k6 = fminf(fmaxf(w[6], -1.f), 1.f) * f;
    float k7 = fminf(fmaxf(w[7], -1.f), 1.f) * f;
    float k8 = fminf(fmaxf(w[8], -1.f), 1.f) * f;

    // ---- 4 columns x 8 rows per thread, 3-row register sliding window ----
    const int cg    = tid & 63;          // column group: cols [4cg, 4cg+4)
    const int rs    = tid >> 6;          // row subgroup: rows [8rs, 8rs+8)
    const int rbase = rs * 8;            // first local output row

    // Window rows hold cols 4cg-1 .. 4cg+4 (6 floats each).
    float T[6], M[6], B[6];

    #define LOADROW(rw, A)                                                  \
        do {                                                                \
            const float* rp = &tile[(rw) * WIDTH + cg * 4];                 \
            float4 v = *reinterpret_cast<const float4*>(rp);                \
            A[0] = (cg == 0)  ? 0.f : rp[-1];                               \
            A[1] = v.x; A[2] = v.y; A[3] = v.z; A[4] = v.w;                 \
            A[5] = (cg == 63) ? 0.f : rp[4];                                \
        } while (0)

    LOADROW(rbase + 0, T);   // LDS row ry   == global row (row0+ry-1)
    LOADROW(rbase + 1, M);

    float* op = out + ((size_t)ch << 16) + (size_t)(row0 + rbase) * WIDTH
              + (size_t)cg * 4;

    #pragma unroll
    for (int i = 0; i < 8; ++i) {
        LOADROW(rbase + i + 2, B);

        float4 o;
        float oo[4];
        #pragma unroll
        for (int j = 0; j < 4; ++j) {
            // shared diagonal partial (identical in conv(k) and conv(k^T))
            float d = k0 * T[j];
            d = fmaf(k4, M[j + 1], d);
            d = fmaf(k8, B[j + 2], d);
            // conv with k
            float a = fmaf(k1, T[j + 1], d);
            a = fmaf(k2, T[j + 2], a);
            a = fmaf(k3, M[j],     a);
            a = fmaf(k5, M[j + 2], a);
            a = fmaf(k6, B[j],     a);
            a = fmaf(k7, B[j + 1], a);
            // conv with k^T
            float b = fmaf(k3, T[j + 1], d);
            b = fmaf(k6, T[j + 2], b);
            b = fmaf(k1, M[j],     b);
            b = fmaf(k7, M[j + 2], b);
            b = fmaf(k2, B[j],     b);
            b = fmaf(k5, B[j + 1], b);
            oo[j] = fabsf(a) + fabsf(b);   // v_add_f32 |a|, |b|
        }
        o.x = oo[0]; o.y = oo[1]; o.z = oo[2]; o.w = oo[3];
        *reinterpret_cast<float4*>(op) = o;  // global_store_b128, coalesced
        op += WIDTH;

        #pragma unroll
        for (int q = 0; q < 6; ++q) { T[q] = M[q]; M[q] = B[q]; }
    }
    #undef LOADROW
}

extern "C" void kernel_launch(void* const* d_in, const int* in_sizes, int n_in,
                              void* d_out, int out_size, void* d_ws, size_t ws_size,
                              hipStream_t stream) {
    (void)in_sizes; (void)n_in; (void)d_ws; (void)ws_size; (void)out_size;
    const float* x  = (const float*)d_in[0];   // 256*256*256
    const float* w  = (const float*)d_in[1];   // 9
    const float* wf = (const float*)d_in[2];   // 1
    float* out = (float*)d_out;

    dim3 grid(NROWS / TH, NCH, 1);   // (8, 256)
    dim3 block(256, 1, 1);           // 8 wave32s
    sobel_g_kernel<<<grid, block, 0, stream>>>(x, w, wf, out);
}